// Block_3822520893918
// MI455X (gfx1250) — compile-verified
//
#include <hip/hip_runtime.h>

// ---------------- constants ----------------
#define NPTS  16384
#define CCH   128
#define NH    32
#define NK    43
#define NKP   44          // padded K for WMMA (43 -> 44, multiple of 4)
#define NC    (NPTS * CCH)

typedef __attribute__((ext_vector_type(2))) float v2f;
typedef __attribute__((ext_vector_type(8))) float v8f;

// =========================================================================
// MT19937 + legacy numpy polar-method gaussian: regenerate kernel points
// (reference builds them with np.random.RandomState(42), they are NOT inputs)
// =========================================================================
struct MTState { unsigned mt[624]; int mti; };

__device__ static void mt_init(MTState& s, unsigned seed) {
    s.mt[0] = seed;
    for (int i = 1; i < 624; ++i)
        s.mt[i] = 1812433253u * (s.mt[i - 1] ^ (s.mt[i - 1] >> 30)) + (unsigned)i;
    s.mti = 624;
}
__device__ static unsigned mt_next(MTState& s) {
    if (s.mti >= 624) {
        for (int i = 0; i < 624; ++i) {
            unsigned y = (s.mt[i] & 0x80000000u) | (s.mt[(i + 1) % 624] & 0x7fffffffu);
            s.mt[i] = s.mt[(i + 397) % 624] ^ (y >> 1);
            if (y & 1u) s.mt[i] ^= 2567483615u;
        }
        s.mti = 0;
    }
    unsigned y = s.mt[s.mti++];
    y ^= y >> 11; y ^= (y << 7) & 2636928640u; y ^= (y << 15) & 4022730752u; y ^= y >> 18;
    return y;
}
__device__ static double mt_double(MTState& s) {
    unsigned a = mt_next(s) >> 5, b = mt_next(s) >> 6;
    return ((double)a * 67108864.0 + (double)b) / 9007199254740992.0;
}
__device__ static double mt_gauss(MTState& s, double& cache, int& has) {
    if (has) { has = 0; return cache; }
    double x1, x2, r2;
    do {
        x1 = 2.0 * mt_double(s) - 1.0;
        x2 = 2.0 * mt_double(s) - 1.0;
        r2 = x1 * x1 + x2 * x2;
    } while (r2 >= 1.0 || r2 == 0.0);
    double f = sqrt(-2.0 * log(r2) / r2);
    cache = f * x1; has = 1;
    return f * x2;
}

__global__ void init_kp_kernel(float* kp /* NK*3 */) {
    if (threadIdx.x != 0 || blockIdx.x != 0) return;
    MTState s; mt_init(s, 42u);
    double cache = 0.0; int has = 0;
    kp[0] = 0.f; kp[1] = 0.f; kp[2] = 0.f;           // shell 0: 1 center point
    const int   sizes[2] = {14, 28};
    const float radii[2] = {0.5f, 1.0f};             // linspace(0, 1, 3)[1:]
    int row = 1;
    for (int sh = 0; sh < 2; ++sh) {
        for (int p = 0; p < sizes[sh]; ++p, ++row) {
            double d0 = mt_gauss(s, cache, has);
            double d1 = mt_gauss(s, cache, has);
            double d2 = mt_gauss(s, cache, has);
            double nrm = sqrt(d0 * d0 + d1 * d1 + d2 * d2);
            kp[row * 3 + 0] = (float)(d0 / nrm) * radii[sh];
            kp[row * 3 + 1] = (float)(d1 / nrm) * radii[sh];
            kp[row * 3 + 2] = (float)(d2 / nrm) * radii[sh];
        }
    }
}

// cwT[c][k] = conv_w[k][c], k padded 43->44 with zero (contiguous-k for v2f B loads)
__global__ void build_cwt_kernel(const float* __restrict__ conv_w, float* __restrict__ cwT) {
    int i = blockIdx.x * blockDim.x + threadIdx.x;
    if (i >= CCH * NKP) return;
    int c = i / NKP, k = i % NKP;
    cwT[i] = (k < NK) ? conv_w[k * CCH + c] : 0.f;
}

// =========================================================================
// GEMM: Y[n][j] = sum_k X[n][k] * W[j][k]   (X: N x 128, W: 128 x 128)
// f32 WMMA 16x16x4; one wave owns a 16-row stripe and all 8 column tiles.
// =========================================================================
__global__ __launch_bounds__(256) void gemm_xwT_kernel(const float* __restrict__ X,
                                                       const float* __restrict__ W,
                                                       float* __restrict__ Y) {
    const int lane  = threadIdx.x & 31;
    const int wave  = threadIdx.x >> 5;                 // 8 waves / block
    const int rbase = (blockIdx.x * 8 + wave) * 16;     // 16 rows per wave
    const int m     = lane & 15;
    const int kh    = lane >> 4;                        // k-pair select (A/B layout)

    v8f acc[8] = {};
    const float* xrow = X + (size_t)(rbase + m) * CCH;

    for (int k0 = 0; k0 < CCH; k0 += 4) {
        v2f a = *(const v2f*)(xrow + k0 + 2 * kh);      // A: 16x4 tile
        __builtin_prefetch(xrow + k0 + 32, 0, 0);
#pragma unroll
        for (int ct = 0; ct < 8; ++ct) {
            int c = ct * 16 + m;                        // B column = out channel
            v2f b = *(const v2f*)(W + (size_t)c * CCH + k0 + 2 * kh);
            acc[ct] = __builtin_amdgcn_wmma_f32_16x16x4_f32(
                false, a, false, b, (short)0, acc[ct], false, false);
        }
    }
#pragma unroll
    for (int ct = 0; ct < 8; ++ct)
#pragma unroll
        for (int r = 0; r < 8; ++r)
            Y[(size_t)(rbase + r + 8 * kh) * CCH + ct * 16 + m] = acc[ct][r];
}

// =========================================================================
// Depthwise KPConv: one block (64 thr = 2 waves) per point.
//   infl (32 x 44) in LDS -> WMMA f32 16x16x4 vs cwT (44 x 128)
//   -> elementwise * gathered neighbor feats -> reduce over h
// =========================================================================
__global__ __launch_bounds__(64) void kpconv_kernel(const float* __restrict__ coord,
                                                    const int*   __restrict__ ridx,
                                                    const float* __restrict__ x1,
                                                    const float* __restrict__ cwT,
                                                    const float* __restrict__ kp,
                                                    float* __restrict__ x2) {
    __shared__ float sI[NH * NKP];      // influence matrix, K padded to 44
    __shared__ float snp[NH][3];        // neighbor - center positions
    __shared__ int   sidx[NH];
    __shared__ float sout[CCH];

    const int n   = blockIdx.x;
    const int tid = threadIdx.x;

    float cx = coord[n * 3 + 0], cy = coord[n * 3 + 1], cz = coord[n * 3 + 2];
    if (tid < NH) {
        int idx = ridx[(size_t)n * NH + tid];
        sidx[tid] = idx;
        float px = 1e6f, py = 1e6f, pz = 1e6f;         // shadow point
        if (idx >= 0 && idx < NPTS) {
            px = coord[idx * 3 + 0]; py = coord[idx * 3 + 1]; pz = coord[idx * 3 + 2];
        }
        snp[tid][0] = px - cx; snp[tid][1] = py - cy; snp[tid][2] = pz - cz;
    }
    sout[tid] = 0.f; sout[tid + 64] = 0.f;
    __syncthreads();

    // linear influence: clip(1 - ||neighbor - kp|| / sigma, 0), sigma = 1
    for (int e = tid; e < NH * NKP; e += 64) {
        int h = e / NKP, k = e % NKP;
        float v = 0.f;
        if (k < NK) {
            float dx = snp[h][0] - kp[k * 3 + 0];
            float dy = snp[h][1] - kp[k * 3 + 1];
            float dz = snp[h][2] - kp[k * 3 + 2];
            v = fmaxf(1.f - sqrtf(dx * dx + dy * dy + dz * dz), 0.f);
        }
        sI[h * NKP + k] = v;
    }
    __syncthreads();

    const int wave = tid >> 5;          // h-tile (0: h 0..15, 1: h 16..31)
    const int lane = tid & 31;
    const int m    = lane & 15;
    const int kh   = lane >> 4;

    v8f acc[8] = {};
    const float* Ih = &sI[(wave * 16 + m) * NKP];
    for (int k0 = 0; k0 < NKP; k0 += 4) {
        v2f a = *(const v2f*)(Ih + k0 + 2 * kh);
#pragma unroll
        for (int ct = 0; ct < 8; ++ct) {
            int c = ct * 16 + m;
            v2f b = *(const v2f*)(cwT + (size_t)c * NKP + k0 + 2 * kh);
            acc[ct] = __builtin_amdgcn_wmma_f32_16x16x4_f32(
                false, a, false, b, (short)0, acc[ct], false, false);
        }
    }

    // W[h][c] * nfeat[h][c], reduce over h (C/D layout: lane l has rows kh*8+r of col m)
#pragma unroll
    for (int ct = 0; ct < 8; ++ct) {
        int c = ct * 16 + m;
        float p = 0.f;
#pragma unroll
        for (int r = 0; r < 8; ++r) {
            int h   = wave * 16 + 8 * kh + r;
            int idx = sidx[h];
            float nf = (idx >= 0 && idx < NPTS) ? x1[(size_t)idx * CCH + c] : 0.f;
            p += acc[ct][r] * nf;
        }
        p += __shfl_xor(p, 16, 32);                    // join row halves
        if (lane < 16) atomicAdd(&sout[c], p);         // join the two h-tile waves
    }
    __syncthreads();
    x2[(size_t)n * CCH + tid]      = sout[tid];
    x2[(size_t)n * CCH + 64 + tid] = sout[64 + tid];
}

// =========================================================================
// BatchNorm statistics: one block per channel
// =========================================================================
__global__ __launch_bounds__(256) void bn_stats_kernel(const float* __restrict__ y,
                                                       float* __restrict__ mean,
                                                       float* __restrict__ rstd) {
    __shared__ float ss[256], sq[256];
    const int c = blockIdx.x;
    float s = 0.f, q = 0.f;
    for (int i = threadIdx.x; i < NPTS; i += 256) {
        float v = y[(size_t)i * CCH + c];
        s += v; q += v * v;
    }
    ss[threadIdx.x] = s; sq[threadIdx.x] = q;
    __syncthreads();
    for (int o = 128; o > 0; o >>= 1) {
        if (threadIdx.x < o) { ss[threadIdx.x] += ss[threadIdx.x + o]; sq[threadIdx.x] += sq[threadIdx.x + o]; }
        __syncthreads();
    }
    if (threadIdx.x == 0) {
        float mu  = ss[0] / (float)NPTS;
        float var = sq[0] / (float)NPTS - mu * mu;     // biased, matches reference
        mean[c] = mu;
        rstd[c] = rsqrtf(var + 1e-5f);
    }
}

// BN apply (+ optional ReLU)
__global__ void bn_apply_kernel(const float* __restrict__ y,
                                const float* __restrict__ mean, const float* __restrict__ rstd,
                                const float* __restrict__ g, const float* __restrict__ b,
                                float* __restrict__ out, int relu) {
    size_t i = (size_t)blockIdx.x * blockDim.x + threadIdx.x;
    if (i >= (size_t)NC) return;
    int c = (int)(i & (CCH - 1));
    float v = (y[i] - mean[c]) * rstd[c] * g[c] + b[c];
    out[i] = relu ? fmaxf(v, 0.f) : v;
}

// final: relu(identity + bn3(y3))
__global__ void final_fuse_kernel(const float* __restrict__ feat, const float* __restrict__ y3,
                                  const float* __restrict__ mean, const float* __restrict__ rstd,
                                  const float* __restrict__ g, const float* __restrict__ b,
                                  float* __restrict__ out_feat) {
    size_t i = (size_t)blockIdx.x * blockDim.x + threadIdx.x;
    if (i >= (size_t)NC) return;
    int c = (int)(i & (CCH - 1));
    float v = (y3[i] - mean[c]) * rstd[c] * g[c] + b[c] + feat[i];
    out_feat[i] = fmaxf(v, 0.f);
}

// pass-through outputs: coord and offset words
__global__ void copy_aux_kernel(const float* __restrict__ coord, const int* __restrict__ offs,
                                float* __restrict__ out, int rem) {
    int i = blockIdx.x * blockDim.x + threadIdx.x;
    if (i < NPTS * 3) out[i] = coord[i];
    if (i < rem) ((int*)(out + NPTS * 3 + (size_t)NC))[i] = offs[i];
}

// =========================================================================
extern "C" void kernel_launch(void* const* d_in, const int* in_sizes, int n_in,
                              void* d_out, int out_size, void* d_ws, size_t ws_size,
                              hipStream_t stream) {
    const float* coord  = (const float*)d_in[0];
    const float* feat   = (const float*)d_in[1];
    const int*   offs   = (const int*)d_in[2];
    const int*   ridx   = (const int*)d_in[3];
    const float* fc1_w  = (const float*)d_in[4];
    const float* fc3_w  = (const float*)d_in[5];
    const float* conv_w = (const float*)d_in[6];
    const float* g1 = (const float*)d_in[7],  *b1 = (const float*)d_in[8];
    const float* g2 = (const float*)d_in[9],  *b2 = (const float*)d_in[10];
    const float* g3 = (const float*)d_in[11], *b3 = (const float*)d_in[12];

    float* ws   = (float*)d_ws;
    float* bufA = ws;                       // NC floats
    float* bufB = ws + (size_t)NC;          // NC floats
    float* kp   = ws + (size_t)2 * NC;      // 132
    float* cwT  = kp + 132;                 // 128*44
    float* mean = cwT + CCH * NKP;          // 128
    float* rstd = mean + CCH;               // 128

    float* out      = (float*)d_out;
    float* out_feat = out + NPTS * 3;

    // init: kernel points + transposed/padded depthwise weights
    init_kp_kernel<<<1, 1, 0, stream>>>(kp);
    build_cwt_kernel<<<(CCH * NKP + 255) / 256, 256, 0, stream>>>(conv_w, cwT);

    const int gemmBlocks = NPTS / (16 * 8);     // 128 rows per block
    const int ncBlocks   = (NC + 255) / 256;

    // fc1 -> BN -> ReLU
    gemm_xwT_kernel<<<gemmBlocks, 256, 0, stream>>>(feat, fc1_w, bufA);
    bn_stats_kernel<<<CCH, 256, 0, stream>>>(bufA, mean, rstd);
    bn_apply_kernel<<<ncBlocks, 256, 0, stream>>>(bufA, mean, rstd, g1, b1, bufB, 1);

    // KPConvD -> BN -> ReLU
    kpconv_kernel<<<NPTS, 64, 0, stream>>>(coord, ridx, bufB, cwT, kp, bufA);
    bn_stats_kernel<<<CCH, 256, 0, stream>>>(bufA, mean, rstd);
    bn_apply_kernel<<<ncBlocks, 256, 0, stream>>>(bufA, mean, rstd, g2, b2, bufB, 1);

    // fc3 -> BN -> residual -> ReLU
    gemm_xwT_kernel<<<gemmBlocks, 256, 0, stream>>>(bufB, fc3_w, bufA);
    bn_stats_kernel<<<CCH, 256, 0, stream>>>(bufA, mean, rstd);
    final_fuse_kernel<<<ncBlocks, 256, 0, stream>>>(feat, bufA, mean, rstd, g3, b3, out_feat);

    // coord + offset pass-through
    int rem = out_size - (NPTS * 3 + NC);
    if (rem < 0) rem = 0;
    if (rem > 2) rem = 2;
    copy_aux_kernel<<<(NPTS * 3 + 255) / 256, 256, 0, stream>>>(coord, offs, out, rem);
}